// GNN_5231270166915
// MI455X (gfx1250) — compile-verified
//
#include <hip/hip_runtime.h>
#include <cmath>

#define N_NODES  100000
#define D        64
#define N_EDGES  1250000
#define N_TILES  (N_NODES / 16)      // 6250, exact
#define WAVES_PER_BLOCK 8

typedef float v2f __attribute__((ext_vector_type(2)));
typedef float v8f __attribute__((ext_vector_type(8)));

// ---------------------------------------------------------------------------
// Kernel 1: zero the aggregation workspace (N*64 f32 in d_ws), float4 stores.
// ---------------------------------------------------------------------------
__global__ __launch_bounds__(256) void sage_zero(float4* __restrict__ aggr4) {
    int i = blockIdx.x * 256 + threadIdx.x;
    if (i < N_NODES * D / 4) {
        aggr4[i] = make_float4(0.f, 0.f, 0.f, 0.f);
    }
}

// ---------------------------------------------------------------------------
// Kernel 2: edge scatter. 16 lanes per edge, each lane handles 4 features:
// coalesced float4 gather of x[src] row (L2-resident), then 4 native
// global_atomic_add_f32 (no return) into aggr[dst] (also L2-resident).
// ---------------------------------------------------------------------------
__global__ __launch_bounds__(256) void sage_scatter(const float* __restrict__ x,
                                                    const long long* __restrict__ edge_index,
                                                    float* __restrict__ aggr) {
    int gid = blockIdx.x * 256 + threadIdx.x;
    int e = gid >> 4;
    if (e >= N_EDGES) return;
    int c = (gid & 15) << 2;                       // feature offset 0,4,...,60
    long long s = edge_index[e];                   // src row (messages)
    long long d = edge_index[N_EDGES + e];         // dst row (aggregation)
    const float4 v = *(const float4*)(x + s * D + c);
    float* base = aggr + d * D + c;
    unsafeAtomicAdd(base + 0, v.x);
    unsafeAtomicAdd(base + 1, v.y);
    unsafeAtomicAdd(base + 2, v.z);
    unsafeAtomicAdd(base + 3, v.w);
}

// ---------------------------------------------------------------------------
// Kernel 3: out = tanh(aggr @ W_l + b_l + x @ W_r) via V_WMMA_F32_16X16X4_F32.
// One wave owns a 16-row node tile and all 64 output columns (4 v8f accums).
//
// Weights staged in LDS pre-swizzled into the WMMA B-fragment layout:
// fragment (kb,nb), lane l, j in {0,1}:
//   LDS float ((kb/4)*4 + nb)*64 + l*2 + j holds W row kb + (l>>4)*2 + j, column nb*16 + (l&15)
// -> every B operand is ONE ds_load_b64 into an aligned VGPR pair (no movs),
//    and 32 lanes x 8B = 256B touches all 64 LDS banks exactly once.
//
// f32 WMMA VGPR layouts (ISA 7.12.2):
//   A (16x4):  M = lane%16, VGPR j holds K = (lane/16)*2 + j
//   B (4x16):  N = lane%16, VGPR j holds K = (lane/16)*2 + j
//   C/D (16x16): VGPR r, lanes 0-15 -> M=r, N=lane; lanes 16-31 -> M=r+8
// ---------------------------------------------------------------------------
__global__ __launch_bounds__(256) void sage_gemm(const float* __restrict__ aggr,
                                                 const float* __restrict__ x,
                                                 const float* __restrict__ W_l,
                                                 const float* __restrict__ b_l,
                                                 const float* __restrict__ W_r,
                                                 float* __restrict__ out) {
    __shared__ float sWl[D * D];   // packed B fragments of W_l (16 KB)
    __shared__ float sWr[D * D];   // packed B fragments of W_r (16 KB)

    // Stage + swizzle: coalesced global read, permuted LDS write (once/block).
    for (int idx = threadIdx.x; idx < D * D; idx += 256) {
        int k = idx >> 6;                           // row of W
        int n = idx & 63;                           // col of W
        int laneP = ((k & 2) << 3) + (n & 15);      // ((k&2)>>1)*16 + n%16
        int dst = (((k >> 2) << 2) + (n >> 4)) * 64 + laneP * 2 + (k & 1);
        sWl[dst] = W_l[idx];
        sWr[dst] = W_r[idx];
    }
    __syncthreads();

    const int wave = threadIdx.x >> 5;
    const int lane = threadIdx.x & 31;
    const int tile = blockIdx.x * WAVES_PER_BLOCK + wave;
    if (tile >= N_TILES) return;                   // wave-uniform: EXEC stays all-1s

    const int m  = lane & 15;                      // M for A fragments
    const int kh = (lane >> 4) << 1;               // K sub-offset: 0 or 2

    const float* aRow = aggr + (tile * 16 + m) * D;
    const float* xRow = x    + (tile * 16 + m) * D;
    const v2f* blFrag = (const v2f*)sWl;           // fragment index = blk*32 + lane
    const v2f* brFrag = (const v2f*)sWr;

    v8f acc[4];
    acc[0] = v8f{}; acc[1] = v8f{}; acc[2] = v8f{}; acc[3] = v8f{};

    #pragma unroll
    for (int kb = 0; kb < 16; ++kb) {              // K-block of 4
        // A fragments: 2 consecutive f32 at K = kb*4 + kh (+0,+1). 8B aligned.
        v2f a_l = *(const v2f*)(aRow + kb * 4 + kh);
        v2f a_r = *(const v2f*)(xRow + kb * 4 + kh);
        #pragma unroll
        for (int nb = 0; nb < 4; ++nb) {
            v2f b_lv = blFrag[(kb * 4 + nb) * 32 + lane];  // one ds_load_b64
            v2f b_rv = brFrag[(kb * 4 + nb) * 32 + lane];
            acc[nb] = __builtin_amdgcn_wmma_f32_16x16x4_f32(
                false, a_l, false, b_lv, (short)0, acc[nb], false, false);
            acc[nb] = __builtin_amdgcn_wmma_f32_16x16x4_f32(
                false, a_r, false, b_rv, (short)0, acc[nb], false, false);
        }
    }

    // Epilogue: bias + tanh, write D-matrix per ISA C/D layout.
    const int rowBase = tile * 16;
    const int mOff    = (lane >> 4) << 3;          // 0 or 8
    #pragma unroll
    for (int nb = 0; nb < 4; ++nb) {
        const int nn = nb * 16 + (lane & 15);
        const float bias = b_l[nn];
        #pragma unroll
        for (int r = 0; r < 8; ++r) {
            const int mm = rowBase + r + mOff;
            out[mm * D + nn] = tanhf(acc[nb][r] + bias);
        }
    }
}

// ---------------------------------------------------------------------------
extern "C" void kernel_launch(void* const* d_in, const int* in_sizes, int n_in,
                              void* d_out, int out_size, void* d_ws, size_t ws_size,
                              hipStream_t stream) {
    const float*      x    = (const float*)d_in[0];
    const long long*  ei   = (const long long*)d_in[1];   // int64 edge_index [2, E]
    const float*      W_l  = (const float*)d_in[2];
    const float*      b_l  = (const float*)d_in[3];
    const float*      W_r  = (const float*)d_in[4];
    float*            out  = (float*)d_out;
    float*            aggr = (float*)d_ws;                // N_NODES * 64 f32 scratch

    (void)in_sizes; (void)n_in; (void)out_size; (void)ws_size;

    // 1) aggr = 0
    {
        int nElem4 = N_NODES * D / 4;
        int blocks = (nElem4 + 255) / 256;
        sage_zero<<<blocks, 256, 0, stream>>>((float4*)aggr);
    }
    // 2) aggr[dst] += x[src]  (16 lanes per edge)
    {
        long long threads = (long long)N_EDGES * 16;
        int blocks = (int)((threads + 255) / 256);
        sage_scatter<<<blocks, 256, 0, stream>>>(x, ei, aggr);
    }
    // 3) out = tanh(aggr @ W_l + b_l + x @ W_r)
    {
        int blocks = (N_TILES + WAVES_PER_BLOCK - 1) / WAVES_PER_BLOCK;
        sage_gemm<<<blocks, 256, 0, stream>>>(aggr, x, W_l, b_l, W_r, out);
    }
}